// AttentionBlock_60000693125476
// MI455X (gfx1250) — compile-verified
//
#include <hip/hip_runtime.h>
#include <math.h>

typedef __attribute__((ext_vector_type(2))) float v2f;
typedef __attribute__((ext_vector_type(8))) float v8f;

#define NEG_SLOPE 0.01f

// ---------------------------------------------------------------------------
// Kernel 1: fused TimeBlock (3x conv1x3 via V_WMMA_F32_16X16X4_F32) + the two
// attention projections si/sj. One workgroup (128 thr = 4 wave32) per (b,n).
// GEMM view per (b,n): A = x-patches (62x48, kk = k*16+ci -> contiguous in X),
// B = stacked conv weights (48x12, cols 12..15 zero-padded), D = (62x12).
// ---------------------------------------------------------------------------
__global__ __launch_bounds__(128) void tb_proj_kernel(
    const float* __restrict__ X,    // (B,N,64,16)
    const float* __restrict__ cw1, const float* __restrict__ cb1,
    const float* __restrict__ cw2, const float* __restrict__ cb2,
    const float* __restrict__ cw3, const float* __restrict__ cb3,
    const float* __restrict__ fcw,  // (2*248)
    float* __restrict__ Si, float* __restrict__ Sj) // (B*N) each
{
    __shared__ float Bmat[48][16];   // weight operand
    __shared__ float Dt[64][16];     // conv results, rows = to, cols = 3*4 outputs
    __shared__ float redI[128];
    __shared__ float redJ[128];

    const int row  = blockIdx.x;     // b*N + n
    const int tid  = threadIdx.x;
    const int wave = tid >> 5;
    const int lane = tid & 31;
    const int hi   = lane >> 4;      // 0: K even pair base 0; 1: base 2
    const int lo   = lane & 15;

    // Build B operand in LDS: Bmat[k*16+ci][j*4+co] = cw_j[co,ci,0,k]
    for (int e = tid; e < 48 * 16; e += 128) {
        const int kk = e >> 4, n = e & 15;
        const int k = kk >> 4, ci = kk & 15;
        float v = 0.0f;
        if (n < 12) {
            const int j = n >> 2, co = n & 3;
            const int widx = co * 48 + ci * 3 + k;       // (Cout,Cin,1,3) layout
            v = (j == 0) ? cw1[widx] : (j == 1) ? cw2[widx] : cw3[widx];
        }
        Bmat[kk][n] = v;
    }
    __syncthreads();

    const float* Xrow = X + (size_t)row * 64 * 16;
    const int to_m = wave * 16 + lo;                 // output position this lane owns (A rows)
    const int to_c = to_m > 61 ? 61 : to_m;          // clamp: keeps patch reads in-bounds

    // 12 K-steps of 16x16x4 f32 WMMA
    v8f acc = {};
    #pragma unroll
    for (int ks = 0; ks < 12; ++ks) {
        const int kkb = 4 * ks + 2 * hi;
        v2f a, b;
        a.x = Xrow[to_c * 16 + kkb];                 // contiguous patch elements
        a.y = Xrow[to_c * 16 + kkb + 1];
        b.x = Bmat[kkb][lo];
        b.y = Bmat[kkb + 1][lo];
        acc = __builtin_amdgcn_wmma_f32_16x16x4_f32(
                  false, a, false, b, (short)0, acc, false, false);
    }

    // D layout: VGPR v -> row M = v + 8*hi, col N = lo
    #pragma unroll
    for (int v = 0; v < 8; ++v)
        Dt[wave * 16 + hi * 8 + v][lo] = acc[v];
    __syncthreads();

    // Activation fusion + projection partials (one thread per valid to)
    float pi = 0.0f, pj = 0.0f;
    if (tid < 62) {
        const int to = tid;
        #pragma unroll
        for (int co = 0; co < 4; ++co) {
            const float y1 = Dt[to][co]     + cb1[co];
            const float y2 = Dt[to][4 + co] + cb2[co];
            const float y3 = Dt[to][8 + co] + cb3[co];
            const float temp = y1 + 1.0f / (1.0f + expf(-y2));   // + sigmoid
            float o = temp + y3;
            o = o > 0.0f ? o : 0.0f;                             // relu
            const int d = to * 4 + co;
            pi += o * fcw[d];
            pj += o * fcw[248 + d];
        }
    }
    redI[tid] = pi;
    redJ[tid] = pj;
    __syncthreads();
    for (int s = 64; s > 0; s >>= 1) {
        if (tid < s) { redI[tid] += redI[tid + s]; redJ[tid] += redJ[tid + s]; }
        __syncthreads();
    }
    if (tid == 0) { Si[row] = redI[0]; Sj[row] = redJ[0]; }
}

// ---------------------------------------------------------------------------
// Kernel 2: masked attention softmax, streaming. One 256-thread block per
// (b,i) row of 2048; 8 elements per lane held in registers; masked entries
// are 0.0 and DO participate in the softmax (matches reference). Output is
// write-once 134 MB -> non-temporal stores. Blocks ordered i-major so the 8
// batches reusing A[i,:] are adjacent (L2 reuse).
// ---------------------------------------------------------------------------
__global__ __launch_bounds__(256) void att_softmax_kernel(
    const float* __restrict__ A,    // (N,N)
    const float* __restrict__ Si, const float* __restrict__ Sj, // (B*N)
    const float* __restrict__ fcb,  // scalar
    float* __restrict__ Out,        // (B,N,N)
    int B, int N)
{
    __shared__ float red[256];
    const int tid = threadIdx.x;
    const int i = blockIdx.x / B;
    const int b = blockIdx.x % B;

    const float siv = Si[(size_t)b * N + i] + fcb[0];
    const float* Arow = A + (size_t)i * N;
    const float* SjB  = Sj + (size_t)b * N;

    float att[8], av[8];
    float m = -INFINITY;
    #pragma unroll
    for (int c = 0; c < 8; ++c) {
        const int j = tid + c * 256;
        const float a = Arow[j];
        const float x = siv + SjB[j];
        const float sc = x > 0.0f ? x : NEG_SLOPE * x;   // leaky_relu
        const float t = (a != 0.0f) ? sc : 0.0f;         // masked -> 0.0
        att[c] = t; av[c] = a;
        m = fmaxf(m, t);
    }
    red[tid] = m;
    __syncthreads();
    for (int s = 128; s > 0; s >>= 1) {
        if (tid < s) red[tid] = fmaxf(red[tid], red[tid + s]);
        __syncthreads();
    }
    m = red[0];
    __syncthreads();

    float e[8];
    float ssum = 0.0f;
    #pragma unroll
    for (int c = 0; c < 8; ++c) { e[c] = expf(att[c] - m); ssum += e[c]; }
    red[tid] = ssum;
    __syncthreads();
    for (int s = 128; s > 0; s >>= 1) {
        if (tid < s) red[tid] += red[tid + s];
        __syncthreads();
    }
    const float inv = 1.0f / red[0];

    float* Orow = Out + ((size_t)b * N + i) * N;
    #pragma unroll
    for (int c = 0; c < 8; ++c) {
        const int j = tid + c * 256;
        __builtin_nontemporal_store(e[c] * inv * av[c], Orow + j);
    }
}

extern "C" void kernel_launch(void* const* d_in, const int* in_sizes, int n_in,
                              void* d_out, int out_size, void* d_ws, size_t ws_size,
                              hipStream_t stream) {
    const float* X   = (const float*)d_in[0];
    const float* A   = (const float*)d_in[1];
    const float* cw1 = (const float*)d_in[2];
    const float* cb1 = (const float*)d_in[3];
    const float* cw2 = (const float*)d_in[4];
    const float* cb2 = (const float*)d_in[5];
    const float* cw3 = (const float*)d_in[6];
    const float* cb3 = (const float*)d_in[7];
    const float* fcw = (const float*)d_in[8];
    const float* fcb = (const float*)d_in[9];
    float* Out = (float*)d_out;

    const int B = 8, N = 2048;
    float* Si = (float*)d_ws;             // B*N floats
    float* Sj = Si + (size_t)B * N;       // B*N floats

    tb_proj_kernel<<<B * N, 128, 0, stream>>>(X, cw1, cb1, cw2, cb2, cw3, cb3,
                                              fcw, Si, Sj);
    att_softmax_kernel<<<(size_t)B * N, 256, 0, stream>>>(A, Si, Sj, fcb, Out, B, N);
}